// PhysicsLoss_12867722018969
// MI455X (gfx1250) — compile-verified
//
#include <hip/hip_runtime.h>

typedef float v2f __attribute__((ext_vector_type(2)));
typedef float v8f __attribute__((ext_vector_type(8)));

#define TLEN 2048
#define NLAG 512
#define PADL 256             // left zero pad (A reads down to -244)
#define XLEN (PADL + 2112)   // right pad: unrolled B reads up to index 2081

// One workgroup (256 threads = 8 wave32) per trajectory sample.
__global__ __launch_bounds__(256) void msd_wmma_kernel(
    const float* __restrict__ alpha_pred,
    const float* __restrict__ D0_pred,
    const float* __restrict__ traj,
    float* __restrict__ ws)
{
    __shared__ float X0[XLEN];     // channel 0, zero-padded
    __shared__ float X1[XLEN];     // channel 1, zero-padded
    __shared__ float cumQ[TLEN];   // inclusive prefix of |r(t)|^2
    __shared__ float red[2048];    // 2 tiles x 4 waves x 256 accum elements
    __shared__ float part[256];    // scan partials / block reduce

    const int b   = blockIdx.x;
    const int tid = threadIdx.x;

    // ---- stage trajectory into padded LDS (pads zeroed) ----
    for (int i = tid; i < XLEN; i += 256) { X0[i] = 0.0f; X1[i] = 0.0f; }
    __syncthreads();
    const float2* tb = (const float2*)(traj + (size_t)b * (size_t)(TLEN * 2));
    #pragma unroll
    for (int k = 0; k < TLEN / 256; ++k) {
        int t = tid + 256 * k;
        float2 p = tb[t];                 // global_load_b64, coalesced
        X0[PADL + t] = p.x;
        X1[PADL + t] = p.y;
    }
    __syncthreads();

    // ---- prefix sum of q(t) = x^2 + y^2 (for the S(l) terms) ----
    float loc[8];
    float run = 0.0f;
    #pragma unroll
    for (int k = 0; k < 8; ++k) {
        int t = tid * 8 + k;
        float x = X0[PADL + t], y = X1[PADL + t];
        run += x * x + y * y;
        loc[k] = run;
    }
    part[tid] = run;
    __syncthreads();
    for (int off = 1; off < 256; off <<= 1) {           // Hillis-Steele scan
        float v = (tid >= off) ? part[tid - off] : 0.0f;
        __syncthreads();
        part[tid] += v;
        __syncthreads();
    }
    float basep = (tid > 0) ? part[tid - 1] : 0.0f;
    float Qtot  = part[255];
    #pragma unroll
    for (int k = 0; k < 8; ++k) cumQ[tid * 8 + k] = basep + loc[k];
    __syncthreads();

    // ---- autocorrelation A(l) via Toeplitz-tiled V_WMMA_F32_16X16X4_F32 ----
    // Tile covers lags l = lagbase + 16*i + j, i,j in 0..15.
    //   a[i,k] = x[w + k - lagbase - 16 i],  b[k,j] = x[w + k + j]
    //   => accumulates x[t] * x[t + l] with t = w + k - lagbase - 16 i.
    // Zero padding makes all out-of-range (t, l) products vanish, so any
    // extra window positions (w >= 2048 from the x2 unroll) contribute 0.
    const int lane = tid & 31;
    // wave-uniform scalars -> force into SGPRs so loop control is scalar
    const int wvs  = __builtin_amdgcn_readfirstlane(tid >> 5);  // wave id 0..7
    const int tile = wvs >> 2;       // lag tile: 0 -> lags 1..256, 1 -> 257..512
    const int wq   = wvs & 3;        // window stripe within tile
    const int lagbase = 1 + 256 * tile;
    const int m = lane & 15;         // M (A) / N (B) index
    const int h = lane >> 4;         // K half: lane holds K = 2h, 2h+1
    const int aOff = PADL - lagbase - 16 * m + 2 * h;
    const int bOff = PADL + m + 2 * h;

    // 4 independent accumulator chains: {channel 0,1} x {window w, w+16}
    v8f accA = {}, accB = {}, accC = {}, accD = {};
    for (int w = lagbase - 3 + 4 * wq; w <= TLEN - 1; w += 32) {
        const int w2 = w + 16;
        v2f a0, b0, a1, b1, a2, b2, a3, b3;
        a0.x = X0[w  + aOff];  a0.y = X0[w  + aOff + 1];
        b0.x = X0[w  + bOff];  b0.y = X0[w  + bOff + 1];
        a1.x = X1[w  + aOff];  a1.y = X1[w  + aOff + 1];
        b1.x = X1[w  + bOff];  b1.y = X1[w  + bOff + 1];
        a2.x = X0[w2 + aOff];  a2.y = X0[w2 + aOff + 1];
        b2.x = X0[w2 + bOff];  b2.y = X0[w2 + bOff + 1];
        a3.x = X1[w2 + aOff];  a3.y = X1[w2 + aOff + 1];
        b3.x = X1[w2 + bOff];  b3.y = X1[w2 + bOff + 1];
        accA = __builtin_amdgcn_wmma_f32_16x16x4_f32(
                   false, a0, false, b0, (short)0, accA, false, false);
        accB = __builtin_amdgcn_wmma_f32_16x16x4_f32(
                   false, a1, false, b1, (short)0, accB, false, false);
        accC = __builtin_amdgcn_wmma_f32_16x16x4_f32(
                   false, a2, false, b2, (short)0, accC, false, false);
        accD = __builtin_amdgcn_wmma_f32_16x16x4_f32(
                   false, a3, false, b3, (short)0, accD, false, false);
    }
    v8f acc = (accA + accC) + (accB + accD);

    // Spill each wave's 16x16 fp32 accumulator tile to LDS.
    // C layout: VGPR r of lane (16h+m) holds element (M = r + 8h, N = m).
    float* rw = &red[tile * 1024 + wq * 256 + lane * 8];
    #pragma unroll
    for (int r = 0; r < 8; ++r) rw[r] = acc[r];
    __syncthreads();

    // ---- per-lag loss; each thread handles 2 lags ----
    const float a_s = alpha_pred[b];
    const float d0  = D0_pred[b];
    float s2 = 0.0f;
    #pragma unroll
    for (int p = 0; p < 2; ++p) {
        int li  = tid + 256 * p;          // 0..511
        int lag = li + 1;
        int tl  = li >> 8;
        int u   = li & 255;               // = 16*M + N within tile
        int M   = u >> 4, N = u & 15;
        int ln  = ((M >> 3) << 4) | N;    // lane that held this element
        int r   = M & 7;                  // VGPR index within accumulator
        const float* rb = &red[tl * 1024 + ln * 8 + r];
        float A = rb[0] + rb[256] + rb[512] + rb[768];   // fixed-order wave sum
        float S = (Qtot - cumQ[lag - 1]) + cumQ[TLEN - 1 - lag];
        float msd  = (S - 2.0f * A) / (float)(TLEN - lag);
        float theo = 4.0f * d0 * powf((float)lag, a_s);
        float d    = logf(theo + 1e-8f) - logf(msd + 1e-8f);
        s2 += d * d;
    }

    // Deterministic block reduction of sum(d^2).
    part[tid] = s2;
    __syncthreads();
    for (int off = 128; off > 0; off >>= 1) {
        if (tid < off) part[tid] += part[tid + off];
        __syncthreads();
    }
    if (tid == 0) ws[b] = part[0] * (1.0f / (float)NLAG);
}

// Fixed-order batch mean (no float atomics -> bit-deterministic).
__global__ __launch_bounds__(256) void msd_finalize_kernel(
    const float* __restrict__ ws, float* __restrict__ out, int n)
{
    __shared__ float sm[256];
    int tid = threadIdx.x;
    float s = 0.0f;
    for (int i = tid; i < n; i += 256) s += ws[i];
    sm[tid] = s;
    __syncthreads();
    for (int off = 128; off > 0; off >>= 1) {
        if (tid < off) sm[tid] += sm[tid + off];
        __syncthreads();
    }
    if (tid == 0) out[0] = sm[0] / (float)n;
}

extern "C" void kernel_launch(void* const* d_in, const int* in_sizes, int n_in,
                              void* d_out, int out_size, void* d_ws, size_t ws_size,
                              hipStream_t stream) {
    const float* alpha = (const float*)d_in[0];   // (B,)
    const float* D0    = (const float*)d_in[1];   // (B,)
    const float* traj  = (const float*)d_in[2];   // (B, T, 2)
    float* out = (float*)d_out;                   // scalar
    float* ws  = (float*)d_ws;                    // B floats of scratch

    const int Bn = in_sizes[0];
    msd_wmma_kernel<<<Bn, 256, 0, stream>>>(alpha, D0, traj, ws);
    msd_finalize_kernel<<<1, 256, 0, stream>>>(ws, out, Bn);
}